// ProgammedController_83485574300001
// MI455X (gfx1250) — compile-verified
//
#include <hip/hip_runtime.h>
#include <math.h>

// ---- problem constants (from reference) ----
#define BATCH   512
#define IF      1024
#define PKEY    64
#define NPROG   32
#define DDIM    131072            // H*W = 512*256
#define MSTRIDE (PKEY + DDIM)     // 131136, row stride of mprogram_weights
#define DELTA   1e-6f

typedef float v2f __attribute__((ext_vector_type(2)));
typedef float v8f __attribute__((ext_vector_type(8)));

// ---------------------------------------------------------------------------
// Kernel 1: MK = tanh(mw[:, :PKEY]); na = ||MK row|| + DELTA      (32x64 tiny)
// one wave, lane p handles program p
// ---------------------------------------------------------------------------
__global__ void pc_prep_programs(const float* __restrict__ mw,
                                 float* __restrict__ MK,
                                 float* __restrict__ na) {
    int p = threadIdx.x;           // 0..31
    float acc = 0.f;
    #pragma unroll 4
    for (int j = 0; j < PKEY; ++j) {
        float v = tanhf(mw[(size_t)p * MSTRIDE + j]);
        MK[p * PKEY + j] = v;
        acc += v * v;
    }
    na[p] = sqrtf(acc) + DELTA;
}

// ---------------------------------------------------------------------------
// Kernel 2: keys = tanh(x @ Wk^T + bk)   (512x1024)@(1024x64)
// f32 WMMA 16x16x4, one wave per 16x16 output tile, K=1024 -> 256 wmma steps
// A layout (16x4 f32): lanes 0-15 hold K=0,1 ; lanes 16-31 hold K=2,3
// C layout: VGPR i -> lanes 0-15: M=i ; lanes 16-31: M=i+8 ; N=lane%16
// ---------------------------------------------------------------------------
__global__ void pc_keys_wmma(const float* __restrict__ x,
                             const float* __restrict__ Wk,
                             const float* __restrict__ bk,
                             float* __restrict__ keys) {
    const int lane = threadIdx.x;          // 32 threads
    const int half = lane >> 4;            // 0 or 1
    const int l15  = lane & 15;
    const int m0 = blockIdx.x * 16;        // batch rows
    const int n0 = blockIdx.y * 16;        // key dims

    v8f acc = {};
    for (int k0 = 0; k0 < IF; k0 += 4) {
        const int kb = k0 + 2 * half;
        v2f a, b;
        a.x = x[(size_t)(m0 + l15) * IF + kb];
        a.y = x[(size_t)(m0 + l15) * IF + kb + 1];
        // B[k][n] = Wk^T[k][n] = Wk[n][k]
        b.x = Wk[(size_t)(n0 + l15) * IF + kb];
        b.y = Wk[(size_t)(n0 + l15) * IF + kb + 1];
        acc = __builtin_amdgcn_wmma_f32_16x16x4_f32(
                  false, a, false, b, (short)0, acc, false, false);
    }
    const int col = n0 + l15;
    const float bias = bk[col];
    #pragma unroll
    for (int i = 0; i < 8; ++i) {
        const int row = m0 + i + 8 * half;
        keys[row * PKEY + col] = tanhf(acc[i] + bias);
    }
}

// ---------------------------------------------------------------------------
// Kernel 3: per-row strength = softplus(x@Ws^T + bs), nb = ||keys row||+DELTA
// one 256-thread block per batch row, LDS tree reductions
// ---------------------------------------------------------------------------
__global__ void pc_row_stats(const float* __restrict__ x,
                             const float* __restrict__ Ws,
                             const float* __restrict__ bs,
                             const float* __restrict__ keys,
                             float* __restrict__ nb,
                             float* __restrict__ strength) {
    __shared__ float sdot[256];
    __shared__ float ssq[256];
    const int b = blockIdx.x;
    const int t = threadIdx.x;

    float d = 0.f;
    #pragma unroll
    for (int j = 0; j < IF / 256; ++j) {
        const int k = t + j * 256;
        d += x[(size_t)b * IF + k] * Ws[k];
    }
    float sq = 0.f;
    if (t < PKEY) {
        float v = keys[b * PKEY + t];
        sq = v * v;
    }
    sdot[t] = d;
    ssq[t] = sq;
    __syncthreads();
    for (int s = 128; s > 0; s >>= 1) {
        if (t < s) {
            sdot[t] += sdot[t + s];
            ssq[t]  += ssq[t + s];
        }
        __syncthreads();
    }
    if (t == 0) {
        float v = sdot[0] + bs[0];
        strength[b] = (v > 20.f) ? v : log1pf(expf(v));   // stable softplus
        nb[b] = sqrtf(ssq[0]) + DELTA;
    }
}

// ---------------------------------------------------------------------------
// Kernel 4: content_weights = softmax(dots/(PKEY*nb*na+DELTA) * strength)
// one wave per batch row, lane p = program p, wave32 shuffle reductions
// ---------------------------------------------------------------------------
__global__ void pc_softmax(const float* __restrict__ keys,
                           const float* __restrict__ MK,
                           const float* __restrict__ na,
                           const float* __restrict__ nb,
                           const float* __restrict__ strength,
                           float* __restrict__ cw) {
    const int b = blockIdx.x;
    const int p = threadIdx.x;   // 0..31

    float dot = 0.f;
    #pragma unroll 4
    for (int j = 0; j < PKEY; ++j)
        dot += keys[b * PKEY + j] * MK[p * PKEY + j];

    const float denom = (float)PKEY * nb[b] * na[p] + DELTA;
    float e = (dot / denom) * strength[b];

    float m = e;
    #pragma unroll
    for (int off = 16; off > 0; off >>= 1)
        m = fmaxf(m, __shfl_xor(m, off, 32));
    float ex = expf(e - m);
    float s = ex;
    #pragma unroll
    for (int off = 16; off > 0; off >>= 1)
        s += __shfl_xor(s, off, 32);
    cw[b * NPROG + p] = ex / s;
}

// ---------------------------------------------------------------------------
// Kernel 5 (the bandwidth-bound one):
//   out = cw(512x32) @ MP(32x131072),  MP[k][n] = mw[k*MSTRIDE + PKEY + n]
// f32 WMMA 16x16x4, K=32 -> 8 wmma steps per 16x16 tile.
// One wave owns 16 rows x 128 cols (8 tiles); A (16x32) held in regs.
// Block = 8 waves = 16 rows x 1024 cols.  Grid = (D/1024, B/16) = (128, 32).
// ---------------------------------------------------------------------------
__global__ void pc_read_wmma(const float* __restrict__ cw,
                             const float* __restrict__ mw,
                             float* __restrict__ out) {
    const int lane = threadIdx.x & 31;
    const int wave = threadIdx.x >> 5;     // 0..7
    const int half = lane >> 4;
    const int l15  = lane & 15;

    const int m0 = blockIdx.y * 16;                      // batch rows
    const int c0 = blockIdx.x * 1024 + wave * 128;       // column base for wave
    const float* __restrict__ MP = mw + PKEY;            // MP[k][n] = MP[k*MSTRIDE + n]

    // Preload A tile: A[m][k] = cw[(m0+m)*32 + k], k-stepped 8x (kb = 4*s + 2*half)
    v2f a[8];
    #pragma unroll
    for (int s = 0; s < 8; ++s) {
        const int kb = 4 * s + 2 * half;
        a[s].x = cw[(m0 + l15) * NPROG + kb];
        a[s].y = cw[(m0 + l15) * NPROG + kb + 1];
    }

    #pragma unroll 2
    for (int t = 0; t < 8; ++t) {
        const int col = c0 + t * 16 + l15;
        if (t < 7) {
            // stream-prefetch next tile's MP rows (global_prefetch_b8)
            __builtin_prefetch(&MP[(size_t)(2 * half) * MSTRIDE + col + 16], 0, 0);
        }
        v8f acc = {};
        #pragma unroll
        for (int s = 0; s < 8; ++s) {
            const int kb = 4 * s + 2 * half;
            v2f bb;
            bb.x = MP[(size_t)kb * MSTRIDE + col];
            bb.y = MP[(size_t)(kb + 1) * MSTRIDE + col];
            acc = __builtin_amdgcn_wmma_f32_16x16x4_f32(
                      false, a[s], false, bb, (short)0, acc, false, false);
        }
        #pragma unroll
        for (int i = 0; i < 8; ++i) {
            const int row = m0 + i + 8 * half;
            out[(size_t)row * DDIM + col] = acc[i];
        }
    }
}

// ---------------------------------------------------------------------------
extern "C" void kernel_launch(void* const* d_in, const int* in_sizes, int n_in,
                              void* d_out, int out_size, void* d_ws, size_t ws_size,
                              hipStream_t stream) {
    const float* x  = (const float*)d_in[0];   // 512 x 1024
    const float* mw = (const float*)d_in[1];   // 32 x 131136
    const float* Wk = (const float*)d_in[2];   // 64 x 1024
    const float* bk = (const float*)d_in[3];   // 64
    const float* Ws = (const float*)d_in[4];   // 1 x 1024
    const float* bs = (const float*)d_in[5];   // 1
    float* out = (float*)d_out;                // 512 x 131072
    float* ws = (float*)d_ws;

    float* keys     = ws;                       // 512*64 = 32768
    float* nb       = keys + BATCH * PKEY;      // 512
    float* strength = nb + BATCH;               // 512
    float* MK       = strength + BATCH;         // 32*64 = 2048
    float* na       = MK + NPROG * PKEY;        // 32
    float* cw       = na + NPROG;               // 512*32 = 16384

    pc_prep_programs<<<1, 32, 0, stream>>>(mw, MK, na);
    pc_keys_wmma<<<dim3(BATCH / 16, PKEY / 16), 32, 0, stream>>>(x, Wk, bk, keys);
    pc_row_stats<<<BATCH, 256, 0, stream>>>(x, Ws, bs, keys, nb, strength);
    pc_softmax<<<BATCH, 32, 0, stream>>>(keys, MK, na, nb, strength, cw);
    pc_read_wmma<<<dim3(DDIM / 1024, BATCH / 16), 256, 0, stream>>>(cw, mw, out);
}